// QuantumLayer_13898514170533
// MI455X (gfx1250) — compile-verified
//
#include <hip/hip_runtime.h>

// CDNA5 / gfx1250 — quantum 4-qubit layer, B = 1M.
// Memory-bound: 16MB in + 16MB out => ~1.4us floor @ 23.3 TB/s.
// Strategy: first RY layer is a rank-1 product built in registers; the
// CNOT chain + RY(weights) layer collapse to one constant 16x16 matrix M
// (built once in d_ws); batched 16x16 matvec done with V_WMMA_F32_16X16X4_F32
// (16 batch rows per wave, K=16 via 4 accumulating WMMAs); Z-expectations via
// a signed Walsh-Hadamard lane butterfly using ds_swizzle, batched across all
// 8 D-VGPRs per stage so the swizzles overlap.

typedef float v2f __attribute__((ext_vector_type(2)));
typedef float v8f __attribute__((ext_vector_type(8)));

// xor-lane shuffle via DS_SWIZZLE (group-of-32 mode): and=0x1F, or=0, xor=d.
#define SWZ_XOR(v, d) \
  __int_as_float(__builtin_amdgcn_ds_swizzle(__float_as_int(v), (((d) << 10) | 0x1F)))

// One FWHT stage over all 8 values: issue 8 independent swizzles, then combine.
#define FWHT_STAGE(d)                                        \
  {                                                          \
    float o0 = SWZ_XOR(p[0], d), o1 = SWZ_XOR(p[1], d);      \
    float o2 = SWZ_XOR(p[2], d), o3 = SWZ_XOR(p[3], d);      \
    float o4 = SWZ_XOR(p[4], d), o5 = SWZ_XOR(p[5], d);      \
    float o6 = SWZ_XOR(p[6], d), o7 = SWZ_XOR(p[7], d);      \
    const bool neg = (n & (d)) != 0;                         \
    p[0] = neg ? (o0 - p[0]) : (p[0] + o0);                  \
    p[1] = neg ? (o1 - p[1]) : (p[1] + o1);                  \
    p[2] = neg ? (o2 - p[2]) : (p[2] + o2);                  \
    p[3] = neg ? (o3 - p[3]) : (p[3] + o3);                  \
    p[4] = neg ? (o4 - p[4]) : (p[4] + o4);                  \
    p[5] = neg ? (o5 - p[5]) : (p[5] + o5);                  \
    p[6] = neg ? (o6 - p[6]) : (p[6] + o6);                  \
    p[7] = neg ? (o7 - p[7]) : (p[7] + o7);                  \
  }

// Build M[i][j] = amplitude i of tail( e_j ), tail = CNOT(0,1);CNOT(1,2);CNOT(2,3);
// then RY(w_q) on each qubit. Qubit q is bit (3-q) of the 4-bit index.
__global__ void build_tail_matrix(const float* __restrict__ w,
                                  float* __restrict__ M) {
  const int t = threadIdx.x;
  if (t >= 256) return;
  const int i = t >> 4;   // output basis index
  const int j = t & 15;   // input basis index
  float c[4], s[4];
#pragma unroll
  for (int q = 0; q < 4; ++q) {
    __sincosf(0.5f * w[q], &s[q], &c[q]);
  }
  // CNOT chain applied to the basis index (classical permutation).
  int jp = j;
  if (jp & 8) jp ^= 4;  // CNOT(q0 -> q1)
  if (jp & 4) jp ^= 2;  // CNOT(q1 -> q2)
  if (jp & 2) jp ^= 1;  // CNOT(q2 -> q3)
  // RY matrix R = [[c, -s], [s, c]] acting per qubit.
  float prod = 1.0f;
#pragma unroll
  for (int q = 0; q < 4; ++q) {
    const int bi = (i >> (3 - q)) & 1;
    const int bj = (jp >> (3 - q)) & 1;
    const float r = bi ? (bj ? c[q] : s[q]) : (bj ? -s[q] : c[q]);
    prod *= r;
  }
  M[i * 16 + j] = prod;
}

__global__ void qlayer_wmma(const float4* __restrict__ x4,
                            const float* __restrict__ M,
                            float* __restrict__ out,
                            int ntiles) {
  const int lane = threadIdx.x & 31;
  const int n    = lane & 15;   // fragment column / batch row within half
  const int kh   = lane >> 4;   // which half of the wave
  const int wid  = blockIdx.x * (blockDim.x >> 5) + (threadIdx.x >> 5);
  const int nw   = gridDim.x * (blockDim.x >> 5);

  // Store-lane mapping, hoisted: after the FWHT, lane 2^(3-q) holds <Z_q>.
  const bool wlane = (n == 1) || (n == 2) || (n == 4) || (n == 8);
  const int  q     = (n == 8) ? 0 : (n == 4) ? 1 : (n == 2) ? 2 : 3;

  // B fragments for D = S * M^T, loaded once (M is 1KB, cache-resident).
  // WMMA B 4x16 layout (wave32): VGPR v holds row K=v (lanes 0-15) and
  // K=v+2 (lanes 16-31), column N = lane%16.  B_j[k,nn] = M[nn, 4j+k].
  v2f bfr[4];
#pragma unroll
  for (int j = 0; j < 4; ++j) {
    bfr[j].x = M[n * 16 + 4 * j + 2 * kh + 0];
    bfr[j].y = M[n * 16 + 4 * j + 2 * kh + 1];
  }

  for (int tile = wid; tile < ntiles; tile += nw) {
    // Lanes L and L+16 both handle batch row tile*16 + (L&15).
    const int row = tile * 16 + n;
    const float4 xv = x4[row];
    float c0, s0, c1, s1, c2, s2, c3, s3;
    __sincosf(0.5f * xv.x, &s0, &c0);
    __sincosf(0.5f * xv.y, &s1, &c1);
    __sincosf(0.5f * xv.z, &s2, &c2);
    __sincosf(0.5f * xv.w, &s3, &c3);
    const float a0v[2] = {c0, s0};
    const float a1v[2] = {c1, s1};
    const float a2 = kh ? s2 : c2;  // this lane's fixed q2 bit

    // A 16x4 layout (wave32): lane<16 -> K = k0+{0,1}; lane>=16 -> K = k0+{2,3}.
    // Global k = 4j + 2*kh + t ; state index bits: b0b1 = j, b2 = kh, b3 = t.
    v8f acc = {0.f, 0.f, 0.f, 0.f, 0.f, 0.f, 0.f, 0.f};
#pragma unroll
    for (int j = 0; j < 4; ++j) {
      const float pj = a0v[j >> 1] * a1v[j & 1] * a2;
      v2f afr;
      afr.x = pj * c3;  // t = 0
      afr.y = pj * s3;  // t = 1
      acc = __builtin_amdgcn_wmma_f32_16x16x4_f32(
          false, afr, false, bfr[j], (short)0, acc, false, false);
    }

    // Epilogue: p = t^2 ; signed 16-lane Walsh butterfly (xor masks <16 keep
    // the two wave halves independent); lane 2^(3-q) ends with <Z_q>.
    float p[8];
#pragma unroll
    for (int v = 0; v < 8; ++v) p[v] = acc[v] * acc[v];
    FWHT_STAGE(1)
    FWHT_STAGE(2)
    FWHT_STAGE(4)
    FWHT_STAGE(8)

    // Single predicated block: lanes {1,2,4,8} (+16) write column q for the
    // 8 batch rows held in this half's D-VGPRs.
    if (wlane) {
      const int rowbase = tile * 16 + 8 * kh;
#pragma unroll
      for (int v = 0; v < 8; ++v) {
        out[(rowbase + v) * 4 + q] = p[v];
      }
    }
  }
}

extern "C" void kernel_launch(void* const* d_in, const int* in_sizes, int n_in,
                              void* d_out, int out_size, void* d_ws, size_t ws_size,
                              hipStream_t stream) {
  const float* x = (const float*)d_in[0];   // [B,4] f32
  const float* w = (const float*)d_in[1];   // [4]   f32
  float* out = (float*)d_out;               // [B,4] f32
  float* M = (float*)d_ws;                  // 256 f32 scratch

  const int B = in_sizes[0] / 4;
  const int ntiles = B / 16;

  hipLaunchKernelGGL(build_tail_matrix, dim3(1), dim3(256), 0, stream, w, M);

  const int blocks = 2048;  // 8 waves/block -> 16384 waves, grid-stride over 65536 tiles
  hipLaunchKernelGGL(qlayer_wmma, dim3(blocks), dim3(256), 0, stream,
                     (const float4*)x, M, out, ntiles);
}